// MultiKR_81982335746122
// MI455X (gfx1250) — compile-verified
//
#include <hip/hip_runtime.h>

// ---------------- problem constants (match reference) ----------------
#define B_TOTAL 16384
#define D 64
#define H1 128
#define H2 64
#define NW 4                       // waves per block
#define ROWS_PER_WAVE 16
#define ROWS_PER_BLOCK (NW * ROWS_PER_WAVE)   // 64

typedef __attribute__((ext_vector_type(2))) float v2f;
typedef __attribute__((ext_vector_type(8))) float v8f;

// D(16x16,f32) = A(16x4,f32) * B(4x16,f32) + C
// A layout: lane L holds row M=L%16, K = 2*(L/16) + {0,1} in the two VGPRs.
// B layout: lane L holds col N=L%16, K = 2*(L/16) + {0,1}.
// C/D layout: lane L holds col N=L%16, rows M = r + 8*(L/16) in VGPR r.
__device__ __forceinline__ v8f wmma_f32_k4(v2f a, v2f b, v8f c) {
  return __builtin_amdgcn_wmma_f32_16x16x4_f32(
      /*neg_a=*/false, a, /*neg_b=*/false, b,
      /*c_mod=*/(short)0, c, /*reuse_a=*/false, /*reuse_b=*/false);
}

__device__ __forceinline__ v8f bcast8(float x) {
  v8f r;
#pragma unroll
  for (int i = 0; i < 8; ++i) r[i] = x;
  return r;
}

__device__ __forceinline__ v8f relu8(v8f x) {
#pragma unroll
  for (int i = 0; i < 8; ++i) x[i] = fmaxf(x[i], 0.0f);
  return x;
}

// 4-way interleaved accumulation: four independent 16x16 tiles (cols
// col0+0/16/32/48) accumulated simultaneously so the XDL pipe sees 4
// independent WMMA chains instead of one serial RAW chain, and each
// B-row slab load feeds 4 WMMAs.
template <int KCH>
__device__ __forceinline__ void gemm_acc4(const v2f* __restrict__ a,
                                          const float* __restrict__ Bm,
                                          int ldb, int col0, int l16, int half,
                                          v8f acc[4]) {
#pragma unroll
  for (int kb = 0; kb < KCH; ++kb) {
    const int k = kb * 4 + half * 2;
    const float* r0 = Bm + (size_t)k * ldb + col0 + l16;
    const float* r1 = r0 + ldb;
    v2f b0, b1, b2, b3;
    b0[0] = r0[0];
    b0[1] = r1[0];
    b1[0] = r0[16];
    b1[1] = r1[16];
    b2[0] = r0[32];
    b2[1] = r1[32];
    b3[0] = r0[48];
    b3[1] = r1[48];
    acc[0] = wmma_f32_k4(a[kb], b0, acc[0]);
    acc[1] = wmma_f32_k4(a[kb], b1, acc[1]);
    acc[2] = wmma_f32_k4(a[kb], b2, acc[2]);
    acc[3] = wmma_f32_k4(a[kb], b3, acc[3]);
  }
}

// Per-row dot product of a C-layout [16 x 64] tile set with a 64-vector.
// x[t][r] = X[m = r + 8*half][n = t*16 + l16]; wv[t] = w[t*16 + l16].
// Reduce across the 16 lanes of each half -> s[r] valid in every lane.
__device__ __forceinline__ void rowdot(const v8f x[4], const float wv[4],
                                       float s[8]) {
#pragma unroll
  for (int r = 0; r < 8; ++r) {
    float p = x[0][r] * wv[0] + x[1][r] * wv[1] + x[2][r] * wv[2] +
              x[3][r] * wv[3];
    p += __shfl_xor(p, 8, 16);
    p += __shfl_xor(p, 4, 16);
    p += __shfl_xor(p, 2, 16);
    p += __shfl_xor(p, 1, 16);
    s[r] = p;
  }
}

__global__ __launch_bounds__(NW * 32) void multikr_fused(
    const int* __restrict__ user_id, const int* __restrict__ item_id,
    const float* __restrict__ rec_target, const float* __restrict__ user_emb,
    const float* __restrict__ item_emb, const float* __restrict__ entity_emb,
    const float* __restrict__ w_vv, const float* __restrict__ w_ev,
    const float* __restrict__ w_ve, const float* __restrict__ w_ee,
    const float* __restrict__ b_v, const float* __restrict__ b_e,
    const float* __restrict__ Wul, const float* __restrict__ bul,
    const float* __restrict__ W1, const float* __restrict__ b1,
    const float* __restrict__ W2, const float* __restrict__ b2,
    const float* __restrict__ W3, const float* __restrict__ b3,
    float* __restrict__ out) {
  __shared__ __attribute__((aligned(16))) float lds[NW * 16 * H1];  // 32 KB

  const int lane = threadIdx.x & 31;
  const int wid = threadIdx.x >> 5;
  const int l16 = lane & 15;
  const int half = lane >> 4;  // 0 or 1
  const int rowBase = (blockIdx.x * NW + wid) * ROWS_PER_WAVE;
  float* myLds = lds + wid * 16 * H1;  // wave-private 16x128 tile

  // ---- second output: rec_target passthrough ----
  if ((int)threadIdx.x < ROWS_PER_BLOCK) {
    const int g = blockIdx.x * ROWS_PER_BLOCK + threadIdx.x;
    out[B_TOTAL + g] = rec_target[g];
  }

  // warm weights into cache (global_prefetch_b8)
  __builtin_prefetch(W1 + (size_t)threadIdx.x * 16, 0, 1);
  __builtin_prefetch(W2 + (size_t)threadIdx.x * 8, 0, 1);

  // ---- gather user rows directly in WMMA A-layout ----
  v2f ua[16];
  {
    const int uid = user_id[rowBase + l16];
    const v2f* up = (const v2f*)(user_emb + (size_t)uid * D);
#pragma unroll
    for (int kb = 0; kb < 16; ++kb) ua[kb] = up[kb * 2 + half];
  }

  // ---- gather item & head (entity) rows in C-layout ----
  v8f itc[4], hdc[4];
#pragma unroll
  for (int r = 0; r < 8; ++r) {
    const int m = r + half * 8;
    const int iid = item_id[rowBase + m];
    const float* ip = item_emb + (size_t)iid * D + l16;
    const float* hp = entity_emb + (size_t)iid * D + l16;
#pragma unroll
    for (int t = 0; t < 4; ++t) {
      itc[t][r] = ip[t * 16];
      hdc[t][r] = hp[t * 16];
    }
  }

  // cross&compress weight vectors (one column each, per-lane slice)
  float wvv[4], wev[4], wve[4], wee[4];
#pragma unroll
  for (int t = 0; t < 4; ++t) {
    wvv[t] = w_vv[t * 16 + l16];
    wev[t] = w_ev[t * 16 + l16];
    wve[t] = w_ve[t * 16 + l16];
    wee[t] = w_ee[t * 16 + l16];
  }
  const float bv = b_v[0], be = b_e[0];

  // ================= 2 layers =================
#pragma unroll
  for (int layer = 0; layer < 2; ++layer) {
    // ---- user = relu(user @ Wul + bul): 4 interleaved 16x16 tiles ----
    {
      v8f acc[4];
#pragma unroll
      for (int j = 0; j < 4; ++j) acc[j] = bcast8(bul[j * 16 + l16]);
      gemm_acc4<16>(ua, Wul, D, 0, l16, half, acc);
#pragma unroll
      for (int j = 0; j < 4; ++j) {
        acc[j] = relu8(acc[j]);
        // C-layout -> row-major LDS (cols 0..63 of the 16x128 "high" tile)
#pragma unroll
        for (int r = 0; r < 8; ++r)
          myLds[(r + half * 8) * H1 + j * 16 + l16] = acc[j][r];
      }
    }
    __syncthreads();
    // reload user as A-layout chunks for the next GEMM
    {
      const v2f* ap = (const v2f*)(myLds + l16 * H1);
#pragma unroll
      for (int kb = 0; kb < 16; ++kb) ua[kb] = ap[kb * 2 + half];
    }
    __syncthreads();

    // ---- cross & compress (per-row matvecs via lane-shuffle reduction) ----
    float s_hvv[8], s_iev[8], s_hve[8], s_iee[8];
    rowdot(hdc, wvv, s_hvv);  // head @ w_vv
    rowdot(itc, wev, s_iev);  // item @ w_ev
    rowdot(hdc, wve, s_hve);  // head @ w_ve
    rowdot(itc, wee, s_iee);  // item @ w_ee
#pragma unroll
    for (int t = 0; t < 4; ++t)
#pragma unroll
      for (int r = 0; r < 8; ++r) {
        const float iv = itc[t][r], hv = hdc[t][r];
        itc[t][r] = iv * s_hvv[r] + hv * s_iev[r] + bv;
        hdc[t][r] = iv * s_hve[r] + hv * s_iee[r] + be;
      }
  }

  // ---- high = [user | item]: item into cols 64..127 (user already there) ----
#pragma unroll
  for (int t = 0; t < 4; ++t)
#pragma unroll
    for (int r = 0; r < 8; ++r)
      myLds[(r + half * 8) * H1 + D + t * 16 + l16] = itc[t][r];
  __syncthreads();

  // ---- hoist all 32 A-chunks of high (16x128) into registers ----
  v2f a32[32];
  {
    const v2f* ap = (const v2f*)(myLds + l16 * H1);
#pragma unroll
    for (int kb = 0; kb < 32; ++kb) a32[kb] = ap[kb * 2 + half];
  }
  __syncthreads();

  // ---- h1 = relu(high @ W1 + b1): 2 groups of 4 interleaved tiles ----
#pragma unroll
  for (int g = 0; g < 2; ++g) {
    v8f acc[4];
#pragma unroll
    for (int j = 0; j < 4; ++j) acc[j] = bcast8(b1[(g * 4 + j) * 16 + l16]);
    gemm_acc4<32>(a32, W1, H1, g * 64, l16, half, acc);
#pragma unroll
    for (int j = 0; j < 4; ++j) {
      acc[j] = relu8(acc[j]);
#pragma unroll
      for (int r = 0; r < 8; ++r)
        myLds[(r + half * 8) * H1 + (g * 4 + j) * 16 + l16] = acc[j][r];
    }
  }
  __syncthreads();

  // ---- reload h1 as A-chunks ----
  {
    const v2f* ap = (const v2f*)(myLds + l16 * H1);
#pragma unroll
    for (int kb = 0; kb < 32; ++kb) a32[kb] = ap[kb * 2 + half];
  }

  // ---- h2 = relu(h1 @ W2 + b2): 4 interleaved tiles, keep in regs ----
  v8f h2[4];
  {
    v8f acc[4];
#pragma unroll
    for (int j = 0; j < 4; ++j) acc[j] = bcast8(b2[j * 16 + l16]);
    gemm_acc4<32>(a32, W2, H2, 0, l16, half, acc);
#pragma unroll
    for (int j = 0; j < 4; ++j) h2[j] = relu8(acc[j]);
  }

  // ---- rec_out = relu(h2 @ W3 + b3): per-row dot + shuffle reduce ----
  float w3v[4];
#pragma unroll
  for (int t = 0; t < 4; ++t) w3v[t] = W3[t * 16 + l16];
  const float b3v = b3[0];
  float s3[8];
  rowdot(h2, w3v, s3);
#pragma unroll
  for (int r = 0; r < 8; ++r) {
    const float v = fmaxf(s3[r] + b3v, 0.0f);
    if (l16 == r) out[rowBase + r + half * 8] = v;  // lanes r and 16+r write
  }
}

extern "C" void kernel_launch(void* const* d_in, const int* in_sizes, int n_in,
                              void* d_out, int out_size, void* d_ws,
                              size_t ws_size, hipStream_t stream) {
  const int* user_id = (const int*)d_in[0];
  const int* item_id = (const int*)d_in[1];
  const float* rec_target = (const float*)d_in[2];
  const float* user_emb = (const float*)d_in[3];
  const float* item_emb = (const float*)d_in[4];
  const float* entity_emb = (const float*)d_in[5];
  const float* w_vv = (const float*)d_in[6];
  const float* w_ev = (const float*)d_in[7];
  const float* w_ve = (const float*)d_in[8];
  const float* w_ee = (const float*)d_in[9];
  const float* b_v = (const float*)d_in[10];
  const float* b_e = (const float*)d_in[11];
  const float* Wul = (const float*)d_in[12];
  const float* bul = (const float*)d_in[13];
  const float* W1 = (const float*)d_in[14];
  const float* b1 = (const float*)d_in[15];
  const float* W2 = (const float*)d_in[16];
  const float* b2 = (const float*)d_in[17];
  const float* W3 = (const float*)d_in[18];
  const float* b3 = (const float*)d_in[19];
  float* out = (float*)d_out;

  dim3 grid(B_TOTAL / ROWS_PER_BLOCK);  // 256 blocks
  dim3 block(NW * 32);                  // 128 threads = 4 waves
  hipLaunchKernelGGL(multikr_fused, grid, block, 0, stream, user_id, item_id,
                     rec_target, user_emb, item_emb, entity_emb, w_vv, w_ev,
                     w_ve, w_ee, b_v, b_e, Wul, bul, W1, b1, W2, b2, W3, b3,
                     out);
}